// SpatialAttention_55233279426566
// MI455X (gfx1250) — compile-verified
//
#include <hip/hip_runtime.h>
#include <math.h>

typedef __attribute__((ext_vector_type(2))) float v2f;
typedef __attribute__((ext_vector_type(8))) float v8f;

#define NB 8
#define NC 256
#define NH 256
#define NW 256
#define HWs (NH * NW)   // 65536 pixels per (b) plane

// ---------------------------------------------------------------------------
// Phase 1: channel-wise mean + max.  One thread = 4 consecutive pixels
// (float4/B128 loads).  Per channel a wave reads 512 contiguous bytes.
// Streams x once (512 MiB), writes 4 MiB of planes into d_ws.
// ---------------------------------------------------------------------------
__global__ __launch_bounds__(256) void k_reduce(const float* __restrict__ x,
                                                float* __restrict__ avg_p,
                                                float* __restrict__ max_p) {
  const int tid = blockIdx.x * 256 + threadIdx.x;   // one float4 of pixels
  const int p4  = tid << 2;                         // pixel index = b*HW + hw
  const int b   = p4 >> 16;
  const int hw  = p4 & (HWs - 1);
  const float4* src = (const float4*)(x + (size_t)b * NC * HWs + hw);
  const size_t cstride4 = HWs / 4;                  // channel stride in float4

  float4 s = make_float4(0.f, 0.f, 0.f, 0.f);
  float4 m = make_float4(-INFINITY, -INFINITY, -INFINITY, -INFINITY);
#pragma unroll 8
  for (int c = 0; c < NC; ++c) {
    float4 v = src[(size_t)c * cstride4];
    s.x += v.x; s.y += v.y; s.z += v.z; s.w += v.w;
    m.x = fmaxf(m.x, v.x); m.y = fmaxf(m.y, v.y);
    m.z = fmaxf(m.z, v.z); m.w = fmaxf(m.w, v.w);
  }
  const float inv = 1.0f / (float)NC;
  ((float4*)avg_p)[tid] = make_float4(s.x * inv, s.y * inv, s.z * inv, s.w * inv);
  ((float4*)max_p)[tid] = m;
}

// ---------------------------------------------------------------------------
// Phase 2: 7x7 conv (2->1 ch) + bias + sigmoid via V_WMMA_F32_16X16X4_F32.
// One wave -> 16 consecutive output pixels in a row.
//   A (16x4): M = pixel-in-tile, K = im2col index (ci*49 + kh*7 + kw, 98 -> 100)
//             ISA layout: lanes 0-15 carry K= kb+0 / kb+1, lanes 16-31 K= kb+2/kb+3
//   B (4x16): weight W[k] broadcast across N -> every column of D is the result.
// Accumulate 25 K-chunks; lanes with N==0 (lane 0 & 16) hold M = acc-idx + 8*(lane>>4).
// EXEC is full at every WMMA (padding handled by per-lane selects).
// ---------------------------------------------------------------------------
__global__ __launch_bounds__(128) void k_conv_wmma(const float* __restrict__ avg_p,
                                                   const float* __restrict__ max_p,
                                                   const float* __restrict__ cw,   // 98 floats
                                                   const float* __restrict__ cb,   // 1 float
                                                   float* __restrict__ attn) {
  const int lane = threadIdx.x & 31;
  const int wave = blockIdx.x * (128 >> 5) + (threadIdx.x >> 5);
  const int p0 = wave << 4;                 // base pixel (16 per wave, same row)
  const int b  = p0 >> 16;
  const int h  = (p0 >> 8) & (NH - 1);
  const int w0 = p0 & (NW - 1);

  const int m    = lane & 15;               // M row handled by this lane (A operand)
  const int koff = (lane >> 4) << 1;        // 0 for lanes 0-15, 2 for lanes 16-31

  const float* __restrict__ pa = avg_p + ((size_t)b << 16);
  const float* __restrict__ pm = max_p + ((size_t)b << 16);

  v8f acc = {0.f, 0.f, 0.f, 0.f, 0.f, 0.f, 0.f, 0.f};

#pragma unroll 5
  for (int kc = 0; kc < 25; ++kc) {
    const int kb = (kc << 2) + koff;
    v2f A, Bv;
#pragma unroll
    for (int j = 0; j < 2; ++j) {
      const int k = kb + j;                 // reduction index, 0..99 (>=98 -> zero pad)
      float av = 0.0f, wv = 0.0f;
      if (k < 98) {
        wv = cw[k];
        const int ci = (k >= 49) ? 1 : 0;
        const int r  = k - ci * 49;
        const int kh = r / 7;
        const int kw = r - kh * 7;
        const int hh = h + kh - 3;
        const int ww = w0 + m + kw - 3;
        if (hh >= 0 && hh < NH && ww >= 0 && ww < NW) {
          const float* __restrict__ pl = ci ? pm : pa;
          av = pl[(hh << 8) + ww];
        }
      }
      A[j]  = av;
      Bv[j] = wv;
    }
    acc = __builtin_amdgcn_wmma_f32_16x16x4_f32(
        /*neg_a=*/false, A, /*neg_b=*/false, Bv,
        /*c_mod=*/(short)0, acc, /*reuse_a=*/false, /*reuse_b=*/false);
  }

  // Column N==0 lanes write the 16 results (8 each), fused bias + sigmoid.
  if (m == 0) {
    const float bias  = cb[0];
    const int   mbase = (lane >> 4) << 3;   // 0 or 8
#pragma unroll
    for (int r = 0; r < 8; ++r) {
      const float z = acc[r] + bias;
      attn[p0 + mbase + r] = 1.0f / (1.0f + __expf(-z));
    }
  }
}

// ---------------------------------------------------------------------------
// Phase 3: out[b,c,h,w] = x[b,c,h,w] * attn[b,h,w].  Pure float4 streaming:
// 512 MiB read + 512 MiB write; the 2 MiB attn plane stays L2-resident.
// ---------------------------------------------------------------------------
__global__ __launch_bounds__(256) void k_apply(const float* __restrict__ x,
                                               const float* __restrict__ attn,
                                               float* __restrict__ out) {
  const long tid = (long)blockIdx.x * 256 + threadIdx.x;
  const long idx = tid << 2;                     // flat element index (b,c,hw)
  const int  hw  = (int)(idx & (HWs - 1));
  const int  b   = (int)(idx >> 24);             // idx / (NC*HWs)
  const float4 xv = ((const float4*)x)[tid];
  const float4 av = *(const float4*)(attn + ((size_t)b << 16) + hw);
  ((float4*)out)[tid] = make_float4(xv.x * av.x, xv.y * av.y,
                                    xv.z * av.z, xv.w * av.w);
}

// ---------------------------------------------------------------------------
extern "C" void kernel_launch(void* const* d_in, const int* in_sizes, int n_in,
                              void* d_out, int out_size, void* d_ws, size_t ws_size,
                              hipStream_t stream) {
  const float* x  = (const float*)d_in[0];   // [8,256,256,256]
  const float* cw = (const float*)d_in[1];   // [1,2,7,7] = 98
  const float* cb = (const float*)d_in[2];   // [1]
  float* out = (float*)d_out;

  float* ws    = (float*)d_ws;               // needs 6 MiB
  float* avg_p = ws;                         // [8*65536]
  float* max_p = ws + (size_t)NB * HWs;      // [8*65536]
  float* attn  = ws + (size_t)2 * NB * HWs;  // [8*65536]

  const int totalPix = NB * HWs;             // 524288

  // Phase 1: 131072 threads (4 px each) -> 512 blocks
  k_reduce<<<totalPix / 4 / 256, 256, 0, stream>>>(x, avg_p, max_p);

  // Phase 2: 32768 waves (16 px each), 4 waves/block -> 8192 blocks
  k_conv_wmma<<<(totalPix / 16) / 4, 128, 0, stream>>>(avg_p, max_p, cw, cb, attn);

  // Phase 3: 33554432 threads (4 el each) -> 131072 blocks
  const long n4 = (long)NB * NC * HWs / 4;
  k_apply<<<(int)(n4 / 256), 256, 0, stream>>>(x, attn, out);
}